// OneHotAndLinear_78675210928791
// MI455X (gfx1250) — compile-verified
//
#include <hip/hip_runtime.h>
#include <hip/hip_bf16.h>
#include <stdint.h>

#define NUM_CLASSES 32000
#define EMBED_DIM   128
#define BT          (64 * 4096)
#define TOK_PER_WAVE 4
#define WAVES_PER_BLK 8

// ---------------------------------------------------------------------------
// gfx1250 async global<->LDS copy builtins (ASYNCcnt path).
// Probe (round 1): builtins exist, arity 4, param0 is
// 'int __vector(4) __device__ *'  ==  v4i addrspace(1)*  (global, 16B vector).
// ---------------------------------------------------------------------------
#if defined(__gfx1250__) && \
    __has_builtin(__builtin_amdgcn_global_load_async_to_lds_b128) && \
    __has_builtin(__builtin_amdgcn_global_store_async_from_lds_b128)
#define HAVE_ASYNC_LDS 1
#else
#define HAVE_ASYNC_LDS 0
#endif

#if HAVE_ASYNC_LDS
typedef int v4i __attribute__((vector_size(16)));
typedef __attribute__((address_space(1))) v4i* gbl_v4i;
typedef __attribute__((address_space(3))) v4i* lds_v4i;

__device__ __forceinline__ gbl_v4i as_global_v4i(const void* p) {
  // flat address == global address numerically; inttoptr avoids const fuss
  return (gbl_v4i)(uintptr_t)p;
}
__device__ __forceinline__ lds_v4i as_lds_v4i(void* p) {
  // real addrspacecast: generic -> LDS offset space
  return (lds_v4i)p;
}

__device__ __forceinline__ void async_wait0() {
#if __has_builtin(__builtin_amdgcn_s_wait_asynccnt)
  __builtin_amdgcn_s_wait_asynccnt(0);
#else
  asm volatile("s_wait_asynccnt 0" ::: "memory");
#endif
}
#endif

// ---------------------------------------------------------------------------
// Pass 2 (placed first so the disasm snippet shows it):
// out[t,:] = Wt[x[t],:]   (bias already folded into Wt)
// One token per wave32: 32 lanes x 16B = 512 B = one embedding row.
// 4 tokens per wave, staged through LDS with the gfx1250 async copy engine:
//   global_load_async_to_lds_b128    (RT: Wt is L2-resident, keep it cached)
//   s_wait_asynccnt 0
//   global_store_async_from_lds_b128 (TH=NT: 134 MB write-once stream)
// ---------------------------------------------------------------------------
__global__ __launch_bounds__(256) void k_gather_async(
    const int* __restrict__ x, const float* __restrict__ Wt,
    float* __restrict__ out) {
#if HAVE_ASYNC_LDS
  __shared__ float stage[WAVES_PER_BLK][TOK_PER_WAVE][EMBED_DIM];  // 16 KB
#endif
  const int lane = threadIdx.x & 31;
  const int wave = threadIdx.x >> 5;
  const int t0 = (blockIdx.x * WAVES_PER_BLK + wave) * TOK_PER_WAVE;

  int idx[TOK_PER_WAVE];
#pragma unroll
  for (int i = 0; i < TOK_PER_WAVE; ++i)
    idx[i] = __builtin_amdgcn_readfirstlane(x[t0 + i]);  // uniform per wave

#if HAVE_ASYNC_LDS
#pragma unroll
  for (int i = 0; i < TOK_PER_WAVE; ++i) {
    const float* src = Wt + (size_t)idx[i] * EMBED_DIM + lane * 4;
    float* sp = &stage[wave][i][lane * 4];
    __builtin_amdgcn_global_load_async_to_lds_b128(
        as_global_v4i(src), as_lds_v4i(sp), /*offset=*/0, /*cpol RT=*/0);
  }
  async_wait0();  // all 4 loads' LDS writes complete (per-wave ASYNCcnt)
#pragma unroll
  for (int i = 0; i < TOK_PER_WAVE; ++i) {
    float* dst = out + (size_t)(t0 + i) * EMBED_DIM + lane * 4;
    float* sp = &stage[wave][i][lane * 4];
    __builtin_amdgcn_global_store_async_from_lds_b128(
        as_global_v4i(dst), as_lds_v4i(sp), /*offset=*/0, /*cpol TH_NT=*/1);
  }
  // stores drain via implicit wait-idle at s_endpgm
#else
#pragma unroll
  for (int i = 0; i < TOK_PER_WAVE; ++i) {
    const float4* src =
        (const float4*)(Wt + (size_t)idx[i] * EMBED_DIM + lane * 4);
    float4* dst = (float4*)(out + (size_t)(t0 + i) * EMBED_DIM + lane * 4);
    *dst = *src;
  }
#endif
}

// ---------------------------------------------------------------------------
// Pass 1: Wt[c][e] = W[e][c] + b[e]   (bias folded; 32x32 LDS tile transpose)
// W: (EMBED_DIM, NUM_CLASSES) row-major -> Wt: (NUM_CLASSES, EMBED_DIM)
// ~33 MB traffic (trivial); makes every gather a contiguous 512 B row, and
// Wt (16.4 MB) stays resident in the 192 MB L2 for the gather pass.
// ---------------------------------------------------------------------------
__global__ __launch_bounds__(256) void k_transpose_bias(
    const float* __restrict__ W, const float* __restrict__ b,
    float* __restrict__ Wt) {
  __shared__ float tile[32][33];  // +1 pad: no LDS bank conflicts
  const int tx = threadIdx.x & 31;
  const int ty = threadIdx.x >> 5;
  const int tile_c = blockIdx.x % (NUM_CLASSES / 32);
  const int tile_e = blockIdx.x / (NUM_CLASSES / 32);
  const int c0 = tile_c * 32;
  const int e0 = tile_e * 32;

#pragma unroll
  for (int r = 0; r < 4; ++r) {
    const int er = ty + 8 * r;                       // coalesced read over c
    tile[er][tx] = W[(size_t)(e0 + er) * NUM_CLASSES + (c0 + tx)] + b[e0 + er];
  }
  __syncthreads();
#pragma unroll
  for (int r = 0; r < 4; ++r) {
    const int cl = ty + 8 * r;                       // coalesced write over e
    Wt[(size_t)(c0 + cl) * EMBED_DIM + (e0 + tx)] = tile[tx][cl];
  }
}

// ---------------------------------------------------------------------------
// Fallback if workspace can't hold Wt (16.4 MB): direct strided column gather.
// ---------------------------------------------------------------------------
__global__ __launch_bounds__(256) void k_gather_direct(
    const int* __restrict__ x, const float* __restrict__ W,
    const float* __restrict__ b, float* __restrict__ out) {
  const int lane = threadIdx.x & 31;
  const int wave = threadIdx.x >> 5;
  const int t = blockIdx.x * WAVES_PER_BLK + wave;
  const int idx = __builtin_amdgcn_readfirstlane(x[t]);
#pragma unroll
  for (int k = 0; k < 4; ++k) {
    const int e = lane + 32 * k;  // coalesced 4B writes per pass
    out[(size_t)t * EMBED_DIM + e] = W[(size_t)e * NUM_CLASSES + idx] + b[e];
  }
}

extern "C" void kernel_launch(void* const* d_in, const int* in_sizes, int n_in,
                              void* d_out, int out_size, void* d_ws,
                              size_t ws_size, hipStream_t stream) {
  const int*   x = (const int*)d_in[0];    // (B,T) class ids
  const float* W = (const float*)d_in[1];  // (EMBED_DIM, NUM_CLASSES)
  const float* b = (const float*)d_in[2];  // (EMBED_DIM,)
  float* out = (float*)d_out;              // (B,T,EMBED_DIM) f32

  const size_t wt_bytes = (size_t)NUM_CLASSES * EMBED_DIM * sizeof(float);
  if (ws_size >= wt_bytes) {
    float* Wt = (float*)d_ws;
    const int n_tiles = (NUM_CLASSES / 32) * (EMBED_DIM / 32);  // 4000
    k_transpose_bias<<<n_tiles, 256, 0, stream>>>(W, b, Wt);
    k_gather_async<<<BT / (WAVES_PER_BLK * TOK_PER_WAVE), 256, 0, stream>>>(
        x, Wt, out);
  } else {
    k_gather_direct<<<BT / WAVES_PER_BLK, 256, 0, stream>>>(x, W, b, out);
  }
}